// LIFSpike_69148973465950
// MI455X (gfx1250) — compile-verified
//
#include <hip/hip_runtime.h>

#ifndef __has_builtin
#define __has_builtin(x) 0
#endif

namespace {
constexpr int   kT      = 100;  // timesteps (reference: last axis)
constexpr int   kT4     = 25;   // kT / 4 -> float4 chunks per sequence
constexpr int   kRows   = 128;  // sequences per block (== blockDim.x)
constexpr int   kPitch4 = 27;   // padded LDS row pitch in float4 (432 B, 16B aligned)
constexpr float kThresh = 0.5f;
constexpr float kDecay  = 0.2f;
constexpr int   kCpolNT = 1;    // TH[2:0] = NT (streaming, no reuse) per ISA §7.3
}

typedef __attribute__((ext_vector_type(4))) float v4f;
typedef __attribute__((ext_vector_type(4))) int   v4i;
typedef __attribute__((address_space(1))) v4i     global_v4i;
typedef __attribute__((address_space(3))) v4i     lds_v4i;

__device__ __forceinline__ void wait_asynccnt0() {
#if __has_builtin(__builtin_amdgcn_s_wait_asynccnt)
    __builtin_amdgcn_s_wait_asynccnt(0);
#else
    asm volatile("s_wait_asynccnt 0" ::: "memory");
#endif
}

// Cooperative, fully coalesced global -> LDS tile copy (lane-linear float4 index
// over the block's contiguous span; LDS destination is the padded [row][col] tile).
template <bool Guard>
__device__ __forceinline__ void load_tile(const v4f* __restrict__ gsrc,
                                          v4f* __restrict__ tile,
                                          int tid, int valid4) {
    #pragma unroll
    for (int k = 0; k < kT4; ++k) {
        const int p = k * kRows + tid;
        if (!Guard || p < valid4) {
            const int row = p / kT4;
            const int col = p - row * kT4;
#if __has_builtin(__builtin_amdgcn_global_load_async_to_lds_b128)
            __builtin_amdgcn_global_load_async_to_lds_b128(
                (global_v4i*)(gsrc + p),
                (lds_v4i*)&tile[row * kPitch4 + col],
                /*offset=*/0, /*cpol=*/kCpolNT);
#else
            tile[row * kPitch4 + col] = gsrc[p];
#endif
        }
    }
}

// Cooperative, fully coalesced LDS -> global tile copy (same mapping).
template <bool Guard>
__device__ __forceinline__ void store_tile(v4f* __restrict__ gdst,
                                           v4f* __restrict__ tile,
                                           int tid, int valid4) {
    #pragma unroll
    for (int k = 0; k < kT4; ++k) {
        const int p = k * kRows + tid;
        if (!Guard || p < valid4) {
            const int row = p / kT4;
            const int col = p - row * kT4;
#if __has_builtin(__builtin_amdgcn_global_store_async_from_lds_b128)
            __builtin_amdgcn_global_store_async_from_lds_b128(
                (global_v4i*)(gdst + p),
                (lds_v4i*)&tile[row * kPitch4 + col],
                /*offset=*/0, /*cpol=*/kCpolNT);
#else
            gdst[p] = tile[row * kPitch4 + col];
#endif
        }
    }
}

__global__ __launch_bounds__(kRows) void lif_scan_kernel(
    const float* __restrict__ x, float* __restrict__ out, int nseq)
{
    // 128 sequences x 27 float4 pitch = 55,296 B of LDS (320 KB/WGP available).
    // Reused in place: holds x on the way in, spikes on the way out.
    __shared__ v4f tile[kRows * kPitch4];

    const int tid = threadIdx.x;
    const long long seq0 = (long long)blockIdx.x * kRows;
    const long long remain = (long long)nseq - seq0;
    const int rowsHere = remain < (long long)kRows ? (int)remain : kRows;
    const int valid4 = rowsHere * kT4;
    const bool fullBlock = (rowsHere == kRows);   // block-uniform (scalar branch)

    // This block's input AND output spans are CONTIGUOUS: kRows * kT floats.
    const v4f* __restrict__ gsrc = (const v4f*)x + seq0 * kT4;
    v4f* __restrict__       gdst = (v4f*)out + seq0 * kT4;

    // ---- Stage 1: coalesced async global -> LDS (ASYNCcnt-tracked) ----
    if (fullBlock) load_tile<false>(gsrc, tile, tid, valid4);
    else           load_tile<true >(gsrc, tile, tid, valid4);
    wait_asynccnt0();      // own wave's async loads done
    __syncthreads();       // all waves' loads done -> tile complete

    // ---- Stage 2: per-thread LIF scan over its private LDS row, spikes
    //      written back in place (own row only -> no cross-thread hazard) ----
    if (tid < rowsHere) {
        v4f* __restrict__ lrow = &tile[tid * kPitch4];
        float m = 0.0f;    // membrane potential
        float s = 0.0f;    // previous spike (0/1)
        #pragma unroll
        for (int k = 0; k < kT4; ++k) {
            v4f xi = lrow[k];              // ds_load_b128 (4 timesteps)
            v4f o;
            #pragma unroll
            for (int j = 0; j < 4; ++j) {
                // mem = mem*DECAY*(1-spike) + x_t  (mul, mul, add order as in reference)
                m = m * kDecay * (1.0f - s) + xi[j];
                s = (m > kThresh) ? 1.0f : 0.0f;
                o[j] = s;
            }
            lrow[k] = o;                   // ds_store_b128 (in-place spike row)
        }
    }
    __syncthreads();       // all rows' spikes committed to LDS

    // ---- Stage 3: coalesced async LDS -> global (ASYNCcnt-tracked) ----
    if (fullBlock) store_tile<false>(gdst, tile, tid, valid4);
    else           store_tile<true >(gdst, tile, tid, valid4);
    wait_asynccnt0();      // explicit; S_ENDPGM would wait-idle anyway
}

extern "C" void kernel_launch(void* const* d_in, const int* in_sizes, int n_in,
                              void* d_out, int out_size, void* d_ws, size_t ws_size,
                              hipStream_t stream)
{
    (void)n_in; (void)out_size; (void)d_ws; (void)ws_size;
    const float* x = (const float*)d_in[0];
    float* out = (float*)d_out;
    const int nseq = in_sizes[0] / kT;                 // 128*4096 = 524288
    const int blocks = (nseq + kRows - 1) / kRows;     // 4096 blocks of 4 wave32s
    lif_scan_kernel<<<blocks, kRows, 0, stream>>>(x, out, nseq);
}